// GCNLayer_49271864819717
// MI455X (gfx1250) — compile-verified
//
#include <hip/hip_runtime.h>
#include <hip/hip_bf16.h>

typedef __attribute__((ext_vector_type(2))) float v2f;
typedef __attribute__((ext_vector_type(8))) float v8f;

#define IN_FEATS  256
#define OUT_FEATS 64

// --------------------------------------------------------------------------
// GEMM: h[n, 64] = x[n, 256] @ W[256, 64] + b[64]
// One wave32 computes a 16-row x 64-col strip of h using four 16x16 f32
// accumulators, iterating K in steps of 4 with V_WMMA_F32_16X16X4_F32.
// A-fragment  (16x4 f32, 2 VGPRs): lane l -> M = l&15, K = 2*(l>>4)+{0,1}
// B-fragment  (4x16 f32, 2 VGPRs): lane l -> N = l&15, K = 2*(l>>4)+{0,1}
// C/D         (16x16 f32, 8 VGPRs): VGPR r, lane l -> M = r+8*(l>>4), N = l&15
// --------------------------------------------------------------------------
__global__ void gcn_gemm_wmma(const float* __restrict__ x,
                              const float* __restrict__ W,
                              const float* __restrict__ bias,
                              float* __restrict__ h,
                              int n_nodes)
{
    const int lane   = threadIdx.x & 31;
    const int waveId = threadIdx.x >> 5;
    const int wavesPerBlock = blockDim.x >> 5;
    const int rowTile = blockIdx.x * wavesPerBlock + waveId;

    const int row0 = rowTile * 16;
    if (row0 >= n_nodes) return;

    const int m  = lane & 15;     // A-row within tile / D-col
    const int hi = lane >> 4;     // 0: K pair {0,1} ; 1: K pair {2,3}
    const int n  = lane & 15;     // B/D column within 16-wide tile

    // Clamp A-row for partial tiles (stores are guarded below).
    int arow = row0 + m;
    if (arow >= n_nodes) arow = n_nodes - 1;
    const float* __restrict__ xrow = x + (size_t)arow * IN_FEATS;

    v8f acc0 = {}, acc1 = {}, acc2 = {}, acc3 = {};

    #pragma unroll 4
    for (int k = 0; k < IN_FEATS; k += 4) {
        const int kk = k + 2 * hi;
        // A fragment: two consecutive f32 from this lane's row of x.
        v2f a = *(const v2f*)(xrow + kk);

        // B fragments for the four 16-column output tiles.
        const float* __restrict__ w0 = W + (size_t)kk * OUT_FEATS + n;       // K = kk
        const float* __restrict__ w1 = w0 + OUT_FEATS;                        // K = kk+1
        v2f b0 = { w0[0],  w1[0]  };
        v2f b1 = { w0[16], w1[16] };
        v2f b2 = { w0[32], w1[32] };
        v2f b3 = { w0[48], w1[48] };

        acc0 = __builtin_amdgcn_wmma_f32_16x16x4_f32(false, a, false, b0, (short)0, acc0, false, false);
        acc1 = __builtin_amdgcn_wmma_f32_16x16x4_f32(false, a, false, b1, (short)0, acc1, false, false);
        acc2 = __builtin_amdgcn_wmma_f32_16x16x4_f32(false, a, false, b2, (short)0, acc2, false, false);
        acc3 = __builtin_amdgcn_wmma_f32_16x16x4_f32(false, a, false, b3, (short)0, acc3, false, false);
    }

    const float bn0 = bias[n];
    const float bn1 = bias[16 + n];
    const float bn2 = bias[32 + n];
    const float bn3 = bias[48 + n];

    #pragma unroll
    for (int r = 0; r < 8; ++r) {
        const int row = row0 + r + 8 * hi;
        if (row < n_nodes) {
            float* __restrict__ hrow = h + (size_t)row * OUT_FEATS;
            hrow[n]      = acc0[r] + bn0;
            hrow[16 + n] = acc1[r] + bn1;
            hrow[32 + n] = acc2[r] + bn2;
            hrow[48 + n] = acc3[r] + bn3;
        }
    }
}

// --------------------------------------------------------------------------
// Edge scatter: out[dst] += w * h[src], 64 feats per edge.
// One wave32 per edge: the edge index is wave-uniform (readfirstlane ->
// SGPR), so src/dst/weight are fetched once per wave on the scalar path.
// Each lane handles 2 features: h[src] is gathered as 32 coalesced b64
// loads (one 256B burst per edge), accumulation via hardware f32 global
// atomics (h and out are L2-resident: 25.6 MB each vs 192 MB L2).
// --------------------------------------------------------------------------
__global__ void gcn_edge_scatter(const float* __restrict__ h,
                                 const int* __restrict__ edge_src,
                                 const int* __restrict__ edge_dst,
                                 const float* __restrict__ edge_w,
                                 float* __restrict__ out,
                                 int n_edges)
{
    const int lane   = threadIdx.x & 31;
    const int waveId = threadIdx.x >> 5;
    const int wavesPerBlock = blockDim.x >> 5;

    // Wave-uniform edge index, forced into an SGPR.
    const int e = __builtin_amdgcn_readfirstlane(blockIdx.x * wavesPerBlock + waveId);
    if (e >= n_edges) return;

    const int   s = edge_src[e];
    const int   d = edge_dst[e];
    const float w = edge_w[e];

    const v2f hv = *(const v2f*)(h + (size_t)s * OUT_FEATS + 2 * lane);

    float* __restrict__ o = out + (size_t)d * OUT_FEATS + 2 * lane;
#if defined(__HIP_DEVICE_COMPILE__)
    unsafeAtomicAdd(o,     w * hv.x);   // global_atomic_add_f32
    unsafeAtomicAdd(o + 1, w * hv.y);
#else
    atomicAdd(o,     w * hv.x);
    atomicAdd(o + 1, w * hv.y);
#endif
}

extern "C" void kernel_launch(void* const* d_in, const int* in_sizes, int n_in,
                              void* d_out, int out_size, void* d_ws, size_t ws_size,
                              hipStream_t stream)
{
    const float* x    = (const float*)d_in[0];
    const int*   esrc = (const int*)  d_in[1];
    const int*   edst = (const int*)  d_in[2];
    const float* ew   = (const float*)d_in[3];
    const float* W    = (const float*)d_in[4];
    const float* bias = (const float*)d_in[5];

    float* out = (float*)d_out;
    float* h   = (float*)d_ws;                     // [n_nodes, 64] scratch

    const int n_nodes = in_sizes[0] / IN_FEATS;    // 100000
    const int n_edges = in_sizes[1];               // 1600000

    // out accumulates via atomics -> must start at zero every call.
    hipMemsetAsync(d_out, 0, (size_t)out_size * sizeof(float), stream);

    // GEMM: one wave per 16-row tile, 8 waves (256 threads) per block.
    const int rowTiles = (n_nodes + 15) / 16;
    const int wavesPerBlock = 8;
    const int gemmBlocks = (rowTiles + wavesPerBlock - 1) / wavesPerBlock;
    gcn_gemm_wmma<<<gemmBlocks, wavesPerBlock * 32, 0, stream>>>(x, W, bias, h, n_nodes);

    // Scatter: one wave32 per edge, 8 waves (256 threads) per block.
    const int scatterBlocks = (n_edges + wavesPerBlock - 1) / wavesPerBlock;
    gcn_edge_scatter<<<scatterBlocks, wavesPerBlock * 32, 0, stream>>>(h, esrc, edst, ew, out, n_edges);
}